// SequenceUnlikelihoodLoss_71992241816094
// MI455X (gfx1250) — compile-verified
//
#include <hip/hip_runtime.h>
#include <hip/hip_bf16.h>
#include <stdint.h>

#define PAD_TOK 50257
#define ALPHA_C 0.2f
#define EPS_C   1e-05f
#define BLK     1024
#define CSZ     2048   // float4s per pipeline chunk (= 2 async b128 per wave)
#define NCHUNK  6

// ---- gfx1250 async global->LDS path (guarded; fallback = plain loads) ----
#if defined(__HIP_DEVICE_COMPILE__) && defined(__gfx1250__) && \
    __has_builtin(__builtin_amdgcn_global_load_async_to_lds_b128) && \
    __has_builtin(__builtin_amdgcn_global_load_async_to_lds_b32)
#define HAVE_ASYNC_LDS 1
#else
#define HAVE_ASYNC_LDS 0
#endif

#if HAVE_ASYNC_LDS
typedef int v4i __attribute__((vector_size(16)));
#define AS1 __attribute__((address_space(1)))
#define AS3 __attribute__((address_space(3)))
// C-style casts: the only cast kind allowed to change the address-space
// qualifier (lowers to addrspacecast).
#define ASYNC_CP_B32(gp, lp) \
  __builtin_amdgcn_global_load_async_to_lds_b32((AS1 int*)(gp), (AS3 int*)(lp), 0, 0)
#define ASYNC_CP_B128(gp, lp) \
  __builtin_amdgcn_global_load_async_to_lds_b128((AS1 v4i*)(gp), (AS3 v4i*)(lp), 0, 0)
#endif

#if defined(__HIP_DEVICE_COMPILE__)
#if __has_builtin(__builtin_amdgcn_s_wait_asynccnt)
#define WAIT_ASYNC(n) do { __builtin_amdgcn_s_wait_asynccnt(n); \
                           asm volatile("" ::: "memory"); } while (0)
#else
#define WAIT_ASYNC(n) asm volatile("s_wait_asynccnt %0" :: "i"(n) : "memory")
#endif
#else
#define WAIT_ASYNC(n) ((void)0)
#endif

// ---- wave32 reductions ----
__device__ __forceinline__ float wred_max(float v) {
#pragma unroll
  for (int o = 16; o > 0; o >>= 1) v = fmaxf(v, __shfl_xor(v, o, 32));
  return v;
}
__device__ __forceinline__ float wred_sum(float v) {
#pragma unroll
  for (int o = 16; o > 0; o >>= 1) v += __shfl_xor(v, o, 32);
  return v;
}

// blockDim.x must be 1024 (32 wave32s)
__device__ __forceinline__ float block_max(float v, float* sc) {
  const int lane = threadIdx.x & 31, wid = threadIdx.x >> 5;
  v = wred_max(v);
  __syncthreads();
  if (lane == 0) sc[wid] = v;
  __syncthreads();
  if (wid == 0) {
    float x = sc[lane];
    x = wred_max(x);
    if (lane == 0) sc[0] = x;
  }
  __syncthreads();
  return sc[0];
}
__device__ __forceinline__ float block_sum(float v, float* sc) {
  const int lane = threadIdx.x & 31, wid = threadIdx.x >> 5;
  v = wred_sum(v);
  __syncthreads();
  if (lane == 0) sc[wid] = v;
  __syncthreads();
  if (wid == 0) {
    float x = sc[lane];
    x = wred_sum(x);
    if (lane == 0) sc[0] = x;
  }
  __syncthreads();
  return sc[0];
}

__device__ __forceinline__ float max4(const float4 v) {
  return fmaxf(fmaxf(v.x, v.y), fmaxf(v.z, v.w));
}
__device__ __forceinline__ float sumexp4(const float4 v, float m) {
  return __expf(v.x - m) + __expf(v.y - m) + __expf(v.z - m) + __expf(v.w - m);
}
// online logsumexp update for one scalar
__device__ __forceinline__ void online_upd(float x, float& m, float& s) {
  if (x > m) { s *= __expf(m - x); m = x; }
  s += __expf(x - m);
}

// ---- kernel 1: firstocc[j] = 1 iff tgt[j] != tgt[j'] for all j' < j ----
__global__ void firstocc_kernel(const int* __restrict__ tgt,
                                int* __restrict__ fo, int N) {
  extern __shared__ int st[];
  for (int k = threadIdx.x; k < N; k += blockDim.x) st[k] = tgt[k];
  __syncthreads();
  const int j = blockIdx.x * blockDim.x + threadIdx.x;
  if (j < N) {
    const int v = st[j];
    int f = 1;
    for (int jp = 0; jp < j; ++jp)
      if (st[jp] == v) { f = 0; break; }
    fo[j] = f;
  }
}

// ---- kernel 2: one workgroup per row. Async-DMA the row into LDS in chunks
//      and compute a single-pass ONLINE logsumexp overlapped with the DMA
//      (per-wave s_wait_asynccnt thresholds; async loads complete in order,
//      and each thread only reads words its own wave issued). Then the mle
//      term and all candidate gathers are served from LDS. ----
__global__ __launch_bounds__(BLK) void row_loss_kernel(
    const float* __restrict__ logits, const int* __restrict__ tgt,
    const int* __restrict__ firstocc, float* __restrict__ partial,
    int N, int V) {
  extern __shared__ __align__(16) float row[];  // V floats + alignment slack
  __shared__ float sc[32];

  const int i = blockIdx.x;
  const int t = threadIdx.x;
  const float* g = logits + (size_t)i * (size_t)V;

  // Align both sides of the b128 copies: shift LDS base by the global
  // residue mod 16 (always a multiple of 4 for fp32 rows).
  const int C = (int)((uintptr_t)g & 15);
  float* rowp = row + (C >> 2);
  const int head  = ((16 - C) & 15) >> 2;          // floats until 16B-aligned
  const int nv4   = (V - head) >> 2;               // aligned float4 count
  const int tail0 = head + (nv4 << 2);

  float lse;
  float m = -3.402823466e+38f;
  float s = 0.f;

#if HAVE_ASYNC_LDS
  float4* rp4 = (float4*)(rowp + head);
  if (nv4 >= NCHUNK * CSZ) {
    // ---- issue: NCHUNK uniform chunks (exactly 2 b128 per wave per chunk,
    //      in order), then remainder, then sub-16B head/tail b32s ----
#pragma unroll
    for (int c = 0; c < NCHUNK; ++c) {
      const int base = c * CSZ;
      ASYNC_CP_B128(g + head + 4 * (size_t)(base + t),       rp4 + base + t);
      ASYNC_CP_B128(g + head + 4 * (size_t)(base + BLK + t), rp4 + base + BLK + t);
    }
    for (int k = NCHUNK * CSZ + t; k < nv4; k += BLK)
      ASYNC_CP_B128(g + head + 4 * (size_t)k, rp4 + k);
    for (int k = t; k < head; k += BLK) ASYNC_CP_B32(g + k, rowp + k);
    for (int k = tail0 + t; k < V; k += BLK) ASYNC_CP_B32(g + k, rowp + k);

    // ---- consume: online logsumexp per chunk, overlapped with DMA.
    //      After chunk c, >= 2*(NCHUNK-1-c) of this wave's ops are still
    //      later than chunk c, so waiting outstanding<=that is safe. ----
#define CHUNK_LSE(c, W)                                         \
    {                                                           \
      WAIT_ASYNC(W);                                            \
      const float4 a = rp4[(c) * CSZ + t];                      \
      const float4 b = rp4[(c) * CSZ + BLK + t];                \
      const float mc = fmaxf(max4(a), max4(b));                 \
      if (mc > m) { s *= __expf(m - mc); m = mc; }              \
      s += sumexp4(a, m) + sumexp4(b, m);                       \
    }
    CHUNK_LSE(0, 10)
    CHUNK_LSE(1, 8)
    CHUNK_LSE(2, 6)
    CHUNK_LSE(3, 4)
    CHUNK_LSE(4, 2)
    CHUNK_LSE(5, 0)
#undef CHUNK_LSE
    WAIT_ASYNC(0);
    // remainder + head/tail (each thread reads only what it issued)
    for (int k = NCHUNK * CSZ + t; k < nv4; k += BLK) {
      const float4 a = rp4[k];
      const float mc = max4(a);
      if (mc > m) { s *= __expf(m - mc); m = mc; }
      s += sumexp4(a, m);
    }
    for (int k = t; k < head; k += BLK) online_upd(rowp[k], m, s);
    for (int k = tail0 + t; k < V; k += BLK) online_upd(rowp[k], m, s);
    __syncthreads();
  } else {
    // generic (small-V) async path: issue all, wait, two-pass
    for (int k = t; k < head; k += BLK) ASYNC_CP_B32(g + k, rowp + k);
    for (int k = t; k < nv4; k += BLK)
      ASYNC_CP_B128(g + head + 4 * (size_t)k, rp4 + k);
    for (int k = tail0 + t; k < V; k += BLK) ASYNC_CP_B32(g + k, rowp + k);
    WAIT_ASYNC(0);
    __syncthreads();
    for (int k = t; k < V; k += BLK) m = fmaxf(m, rowp[k]);
    m = block_max(m, sc);
    for (int k = t; k < V; k += BLK) s += __expf(rowp[k] - m);
    // fold into (m, s) form for the shared combine below
  }
#else
  for (int k = t; k < head; k += BLK) rowp[k] = g[k];
  for (int k = t; k < nv4; k += BLK) {
    const float4 d = *(const float4*)(g + head + 4 * (size_t)k);
    *(float4*)(rowp + head + 4 * k) = d;
  }
  for (int k = tail0 + t; k < V; k += BLK) rowp[k] = g[k];
  __syncthreads();
  for (int k = t; k < V; k += BLK) m = fmaxf(m, rowp[k]);
  m = block_max(m, sc);
  for (int k = t; k < V; k += BLK) s += __expf(rowp[k] - m);
#endif

  // ---- combine per-thread (m, s) -> block logsumexp ----
  const float mstar = block_max(m, sc);
  const float sadj = s * __expf(m - mstar);
  const float S = block_sum(sadj, sc);
  lse = mstar + __logf(S);

  const int ti = tgt[i];

  // candidate gathers from LDS: distinct previous targets != ti, != PAD
  float cacc = 0.f;
  for (int j = t; j < i; j += BLK) {
    const int v = tgt[j];
    if (firstocc[j] && v != ti && v != PAD_TOK) {
      const float p = __expf(rowp[v] - lse);
      const float omp = fmaxf(1.f - p, EPS_C);
      cacc -= __logf(omp);
    }
  }
  cacc = block_sum(cacc, sc);

  if (t == 0) {
    // PAD candidate is present for every row (keep==false branch of scatter)
    const float pP = __expf(rowp[PAD_TOK] - lse);
    const float ompP = fmaxf(1.f - pP, EPS_C);
    const float custom = cacc - __logf(ompP);
    const float mle = (ti == PAD_TOK) ? 0.f : -(rowp[ti] - lse);
    partial[i] = mle + ALPHA_C * custom;
  }
}

// ---- kernel 3: deterministic final reduction ----
__global__ __launch_bounds__(256) void reduce_kernel(
    const float* __restrict__ partial, float* __restrict__ out, int N) {
  __shared__ float sc[8];
  float s = 0.f;
  for (int k = threadIdx.x; k < N; k += 256) s += partial[k];
  s = wred_sum(s);
  const int lane = threadIdx.x & 31, wid = threadIdx.x >> 5;
  if (lane == 0) sc[wid] = s;
  __syncthreads();
  if (threadIdx.x == 0) {
    float tot = 0.f;
    for (int w = 0; w < 8; ++w) tot += sc[w];
    out[0] = tot;
  }
}

extern "C" void kernel_launch(void* const* d_in, const int* in_sizes, int n_in,
                              void* d_out, int out_size, void* d_ws, size_t ws_size,
                              hipStream_t stream) {
  (void)n_in; (void)out_size; (void)ws_size;
  const float* logits  = (const float*)d_in[0];
  const int*   targets = (const int*)d_in[1];
  const int N = in_sizes[1];                                   // B*T = 2048
  const int V = (int)((long long)in_sizes[0] / (long long)N);  // 50258

  float* out     = (float*)d_out;
  float* partial = (float*)d_ws;
  const size_t fo_off = (((size_t)N * sizeof(float)) + 255) & ~(size_t)255;
  int* fo = (int*)((char*)d_ws + fo_off);

  firstocc_kernel<<<(N + 255) / 256, 256, (size_t)N * sizeof(int), stream>>>(
      targets, fo, N);

  const size_t sh = (size_t)V * sizeof(float) + 16;  // row + alignment slack
  row_loss_kernel<<<N, BLK, sh, stream>>>(logits, targets, fo, partial, N, V);

  reduce_kernel<<<1, 256, 0, stream>>>(partial, out, N);
}